// BilinearInterpolation_45414984187976
// MI455X (gfx1250) — compile-verified
//
#include <hip/hip_runtime.h>

typedef __attribute__((ext_vector_type(2))) float v2f;
typedef __attribute__((ext_vector_type(4))) float v4f;
typedef __attribute__((ext_vector_type(8))) float v8f;

#define OUT_H 256
#define OUT_W 256
#define IN_H 256
#define IN_W 256
#define CH 32
#define PIX_PER_BLOCK 16
#define THREADS (PIX_PER_BLOCK * (CH / 4)) // 128 threads = 4 wave32

// Bilinear spatial-transformer sampler.
// grid.x tiles the 65536 output pixels (16 per block), grid.y = batch.
__global__ __launch_bounds__(THREADS) void stn_bilinear_kernel(
    const float* __restrict__ X,  // [B, H, W, C] fp32
    const float* __restrict__ T,  // [B, 6] affine params
    float* __restrict__ out)      // [B, OUT_H, OUT_W, C] fp32
{
    __shared__ float sxs[PIX_PER_BLOCK];
    __shared__ float sys[PIX_PER_BLOCK];

    const int tid = threadIdx.x;
    const int b   = blockIdx.y;              // uniform -> scalar theta loads
    const float* th = T + b * 6;

#if __has_builtin(__builtin_amdgcn_wmma_f32_16x16x4_f32)
    // Wave 0 computes sampled = theta(2x3) @ grid(3x16pixels) via one
    // v_wmma_f32_16x16x4_f32:  D[m][0]=sx_m, D[m][1]=sy_m.
    if (tid < 32) {
        const int lane = tid;
        const int m = lane & 15;
        const int p = blockIdx.x * PIX_PER_BLOCK + m;     // flat output pixel
        const float gx = -1.0f + (float)(p % OUT_W) * (2.0f / (float)(OUT_W - 1));
        const float gy = -1.0f + (float)(p / OUT_W) * (2.0f / (float)(OUT_H - 1));

        // A (16x4 f32): lanes 0-15 hold K=0,1 ; lanes 16-31 hold K=2,3
        v2f a;
        a.x = (lane < 16) ? gx : 1.0f;
        a.y = (lane < 16) ? gy : 0.0f;

        // B (4x16 f32): lane n holds K=0,1 (lanes 0-15) / K=2,3 (lanes 16-31)
        v2f bm; bm.x = 0.0f; bm.y = 0.0f;
        if (lane == 0)       { bm.x = th[0]; bm.y = th[1]; }   // col0 <- theta row0
        else if (lane == 16) { bm.x = th[2];               }
        else if (lane == 1)  { bm.x = th[3]; bm.y = th[4]; }   // col1 <- theta row1
        else if (lane == 17) { bm.x = th[5];               }

        v8f c = {};
        v8f d = __builtin_amdgcn_wmma_f32_16x16x4_f32(
            /*neg_a=*/false, a, /*neg_b=*/false, bm,
            /*c_mod=*/(short)0, c, /*reuse_a=*/false, /*reuse_b=*/false);

        // D layout: d[j] = D[j][lane] (lanes 0-15), D[j+8][lane-16] (16-31)
        float* dst = nullptr;
        if (lane == 0)       dst = &sxs[0];   // sx pixels 0..7
        else if (lane == 16) dst = &sxs[8];   // sx pixels 8..15
        else if (lane == 1)  dst = &sys[0];   // sy pixels 0..7
        else if (lane == 17) dst = &sys[8];   // sy pixels 8..15
        if (dst) {
#pragma unroll
            for (int j = 0; j < 8; ++j) dst[j] = d[j];
        }
    }
#else
    if (tid < PIX_PER_BLOCK) {
        const int p = blockIdx.x * PIX_PER_BLOCK + tid;
        const float gx = -1.0f + (float)(p % OUT_W) * (2.0f / (float)(OUT_W - 1));
        const float gy = -1.0f + (float)(p / OUT_W) * (2.0f / (float)(OUT_H - 1));
        sxs[tid] = fmaf(th[0], gx, fmaf(th[1], gy, th[2]));
        sys[tid] = fmaf(th[3], gx, fmaf(th[4], gy, th[5]));
    }
#endif
    __syncthreads();

    const int pix = tid >> 3;   // pixel within block (0..15)
    const int q   = tid & 7;    // float4 channel chunk (0..7)
    const int p   = blockIdx.x * PIX_PER_BLOCK + pix;

    const float sx = sxs[pix];
    const float sy = sys[pix];
    // Faithful to reference: x scaled by H, y scaled by W.
    const float x = 0.5f * (sx + 1.0f) * (float)IN_H;
    const float y = 0.5f * (sy + 1.0f) * (float)IN_W;

    // astype(int32) truncates toward zero, as does (int) cast.
    int x0 = (int)x, y0 = (int)y;
    int x1 = x0 + 1, y1 = y0 + 1;
    x0 = min(max(x0, 0), IN_H - 1);
    x1 = min(max(x1, 0), IN_H - 1);
    y0 = min(max(y0, 0), IN_W - 1);
    y1 = min(max(y1, 0), IN_W - 1);

    // Weights from the *clipped* integer coords (matches reference).
    const float wx1 = (float)x1 - x, wx0 = x - (float)x0;
    const float wy1 = (float)y1 - y, wy0 = y - (float)y0;
    const float wa = wx1 * wy1, wb = wx1 * wy0;
    const float wc = wx0 * wy1, wd = wx0 * wy0;

    const int basep = b * (IN_H * IN_W);
    const v4f* __restrict__ flat = (const v4f*)X;        // 8 x float4 per pixel
    const int ia = (basep + y0 * IN_W + x0) * (CH / 4) + q;
    const int ib = (basep + y1 * IN_W + x0) * (CH / 4) + q;
    const int ic = (basep + y0 * IN_W + x1) * (CH / 4) + q;
    const int id = (basep + y1 * IN_W + x1) * (CH / 4) + q;

    const v4f pa = flat[ia];   // global_load_b128, RT (keep input in L2)
    const v4f pb = flat[ib];
    const v4f pc = flat[ic];
    const v4f pd = flat[id];

    v4f r;
    r.x = wa * pa.x + wb * pb.x + wc * pc.x + wd * pd.x;
    r.y = wa * pa.y + wb * pb.y + wc * pc.y + wd * pd.y;
    r.z = wa * pa.z + wb * pb.z + wc * pc.z + wd * pd.z;
    r.w = wa * pa.w + wb * pb.w + wc * pc.w + wd * pd.w;

    // Streaming write-once output: non-temporal so it doesn't evict the
    // 134MB input working set from the 192MB L2.
    v4f* optr = (v4f*)out + (basep + p) * (CH / 4) + q;
    __builtin_nontemporal_store(r, optr);
}

extern "C" void kernel_launch(void* const* d_in, const int* in_sizes, int n_in,
                              void* d_out, int out_size, void* d_ws, size_t ws_size,
                              hipStream_t stream) {
    const float* X = (const float*)d_in[0];  // [B,256,256,32] fp32
    const float* T = (const float*)d_in[1];  // [B,6] fp32
    float* out = (float*)d_out;

    const int B = in_sizes[1] / 6;           // 16
    dim3 grid((OUT_H * OUT_W) / PIX_PER_BLOCK, B);
    hipLaunchKernelGGL(stn_bilinear_kernel, grid, dim3(THREADS), 0, stream,
                       X, T, out);
}